// OhemNegLossnew_78915729097126
// MI455X (gfx1250) — compile-verified
//
#include <hip/hip_runtime.h>
#include <hip/hip_bf16.h>
#include <stdint.h>

#define NBINS      4096
#define THREADS    256
#define TILE_ELEMS (THREADS * 4)      // 1024 elements / tile (b128 per lane)
#define NSLOTS     4                  // 4-deep async ring, 3 tiles in flight
#define QSHIFT     20                 // fixed-point scale 2^20
#define CNT_SHIFT  44                 // packed bin: count<<44 | sum_q

__device__ __forceinline__ unsigned lds_addr(const void* p) {
    // generic pointer to __shared__: low 32 bits are the DS byte address
    return (unsigned)(uintptr_t)p;
}

__global__ void __launch_bounds__(THREADS)
ohem_pass1(const float* __restrict__ dp, const float* __restrict__ dt,
           long long N, int tilesPerBlock,
           unsigned long long* __restrict__ hist,
           unsigned long long* __restrict__ posAcc,
           unsigned* __restrict__ posCnt)
{
    __shared__ unsigned long long s_hist[NBINS];         // 32 KB
    __shared__ float s_p[NSLOTS][TILE_ELEMS];            // 16 KB
    __shared__ float s_t[NSLOTS][TILE_ELEMS];            // 16 KB

    const int tid = threadIdx.x;
    for (int b = tid; b < NBINS; b += THREADS) s_hist[b] = 0ULL;
    __syncthreads();

    unsigned long long posQ = 0ULL;
    unsigned posN = 0;

    auto proc = [&](float p, float t) {
        const bool isPos = (t == 1.0f);
        const bool isNeg = (t == 0.0f);
        const float x = isPos ? p : (1.0f - p);
        float l = fmaxf(__logf(x), -100.0f);             // v_log_f32 path
        const unsigned long long q =
            (unsigned long long)(-l * 1048576.0f + 0.5f);
        if (isNeg) {
            int b = (int)(p * (float)NBINS);
            b = b < 0 ? 0 : (b > NBINS - 1 ? NBINS - 1 : b);
            atomicAdd(&s_hist[b], (1ULL << CNT_SHIFT) | q);
        } else if (isPos) {
            posQ += q;
            ++posN;
        }
    };

    const long long numTiles = N / TILE_ELEMS;
    const long long tile0    = (long long)blockIdx.x * tilesPerBlock;
    long long tileEnd = tile0 + tilesPerBlock;
    if (tileEnd > numTiles) tileEnd = numTiles;
    const int nT = (int)(tileEnd > tile0 ? (tileEnd - tile0) : 0);

    // per-lane LDS destinations for the 4-slot ring (hoisted)
    unsigned ldsP[NSLOTS], ldsT[NSLOTS];
#pragma unroll
    for (int s = 0; s < NSLOTS; ++s) {
        ldsP[s] = lds_addr(&s_p[s][tid * 4]);
        ldsT[s] = lds_addr(&s_t[s][tid * 4]);
    }

    auto issue = [&](int t) {
        const int slot = t & (NSLOTS - 1);
        const unsigned goff =
            (unsigned)(((tile0 + t) * (long long)TILE_ELEMS + tid * 4) * 4LL);
        // CDNA5 async DMA: global -> LDS, 16B per lane, tracked by ASYNCcnt
        asm volatile("global_load_async_to_lds_b128 %0, %1, %2 offset:0"
                     :: "v"(ldsP[slot]), "v"(goff), "s"(dp) : "memory");
        asm volatile("global_load_async_to_lds_b128 %0, %1, %2 offset:0"
                     :: "v"(ldsT[slot]), "v"(goff), "s"(dt) : "memory");
    };

    if (nT > 0) {
        // prologue: up to 3 tiles in flight
        issue(0);
        if (1 < nT) issue(1);
        if (2 < nT) issue(2);
        for (int t = 0; t < nT; ++t) {
            const int slot = t & (NSLOTS - 1);
            // keep 3 tiles ahead; drain ladder with literal immediates
            if (t + 3 < nT) {
                issue(t + 3);
                asm volatile("s_wait_asynccnt 0x6" ::: "memory");
            } else if (t + 2 < nT) {
                asm volatile("s_wait_asynccnt 0x4" ::: "memory");
            } else if (t + 1 < nT) {
                asm volatile("s_wait_asynccnt 0x2" ::: "memory");
            } else {
                asm volatile("s_wait_asynccnt 0x0" ::: "memory");
            }
            const float4 p4 = *reinterpret_cast<const float4*>(&s_p[slot][tid * 4]);
            const float4 t4 = *reinterpret_cast<const float4*>(&s_t[slot][tid * 4]);
            proc(p4.x, t4.x);
            proc(p4.y, t4.y);
            proc(p4.z, t4.z);
            proc(p4.w, t4.w);
        }
    }

    // tail (N not multiple of TILE_ELEMS): plain global loads, grid-stride
    const long long tailStart = numTiles * (long long)TILE_ELEMS;
    for (long long i = tailStart + (long long)blockIdx.x * THREADS + tid;
         i < N; i += (long long)gridDim.x * THREADS) {
        proc(dp[i], dt[i]);
    }

    __syncthreads();
    // flush block-local histogram with integer atomics (order-independent)
    for (int b = tid; b < NBINS; b += THREADS) {
        const unsigned long long h = s_hist[b];
        if (h) atomicAdd(&hist[b], h);
    }
    if (posN) {
        atomicAdd(posAcc, posQ);
        atomicAdd(posCnt, posN);
    }
}

__global__ void __launch_bounds__(THREADS)
ohem_pass2(const unsigned long long* __restrict__ hist,
           const unsigned long long* __restrict__ posAcc,
           const unsigned* __restrict__ posCnt,
           const float* __restrict__ lp, const float* __restrict__ lt,
           int nl, float* __restrict__ out)
{
    __shared__ unsigned long long sh[NBINS];   // 32 KB
    __shared__ double red[THREADS];
    const int tid = threadIdx.x;

    for (int b = tid; b < NBINS; b += THREADS) sh[b] = hist[b];

    // MSE over the small label arrays
    double e = 0.0;
    for (int i = tid; i < nl; i += THREADS) {
        const float d = lp[i] - lt[i];
        e += (double)d * (double)d;
    }
    red[tid] = e;
    __syncthreads();
    for (int s = THREADS / 2; s > 0; s >>= 1) {
        if (tid < s) red[tid] += red[tid + s];
        __syncthreads();
    }

    if (tid == 0) {
        const unsigned long long SUM_MASK = (1ULL << CNT_SHIFT) - 1ULL;
        unsigned long long nneg = 0ULL;
        for (int b = 0; b < NBINS; ++b) nneg += (sh[b] >> CNT_SHIFT);

        const long long k = (long long)floorf(0.7f * (float)nneg);

        double sum = 0.0;
        long long need = k;
        for (int b = NBINS - 1; b >= 0 && need > 0; --b) {
            const unsigned long long h = sh[b];
            const long long c = (long long)(h >> CNT_SHIFT);
            if (c == 0) continue;
            const double sq = (double)(h & SUM_MASK);
            if (c <= need) { sum += sq; need -= c; }
            else           { sum += sq * (double)need / (double)c; need = 0; }
        }
        const double mean_neg = (k > 0) ? (sum / 1048576.0) / (double)k : 0.0;

        const unsigned np = *posCnt;
        const double mean_pos =
            (np > 0) ? ((double)(*posAcc) / 1048576.0) / (double)np : 0.0;

        const double mse = red[0] / (double)nl;
        out[0] = (float)(mse + mean_pos + mean_neg);
    }
}

extern "C" void kernel_launch(void* const* d_in, const int* in_sizes, int n_in,
                              void* d_out, int out_size, void* d_ws, size_t ws_size,
                              hipStream_t stream) {
    const float* lp = (const float*)d_in[0];
    const float* lt = (const float*)d_in[1];
    const float* dp = (const float*)d_in[2];
    const float* dt = (const float*)d_in[3];
    const long long N  = (long long)in_sizes[2];
    const int       nl = in_sizes[0];

    unsigned long long* hist   = (unsigned long long*)d_ws;
    unsigned long long* posAcc = hist + NBINS;
    unsigned*           posCnt = (unsigned*)(posAcc + 1);

    // zero accumulators (graph-capture legal)
    hipMemsetAsync(d_ws, 0, (size_t)(NBINS + 1) * 8 + 8, stream);

    const int grid = 512;
    const long long numTiles = N / TILE_ELEMS;
    const int tilesPerBlock = (int)((numTiles + grid - 1) / grid);

    hipLaunchKernelGGL(ohem_pass1, dim3(grid), dim3(THREADS), 0, stream,
                       dp, dt, N, tilesPerBlock, hist, posAcc, posCnt);
    hipLaunchKernelGGL(ohem_pass2, dim3(1), dim3(THREADS), 0, stream,
                       hist, posAcc, posCnt, lp, lt, nl, (float*)d_out);
}